// MultiHeadAttention_9268539425129
// MI455X (gfx1250) — compile-verified
//
#include <hip/hip_runtime.h>

#define D_MODEL 1024
#define NUM_HEADS 16
#define D_K 64
#define BATCH 2
#define SEQ 2048

typedef __attribute__((ext_vector_type(2))) float v2f;
typedef __attribute__((ext_vector_type(8))) float v8f;

// Native fp32 WMMA: D(16x16) = A(16x4) * B(4x16) + C
__device__ __forceinline__ v8f wmma_f32(v2f a, v2f b, v8f c) {
    return __builtin_amdgcn_wmma_f32_16x16x4_f32(
        false, a, false, b, (short)0, c, false, false);
}

// ---------------------------------------------------------------------------
// Pair-pack a row-major [Ktot x N] matrix so that the WMMA B-fragment
// (X[k][n], X[k+1][n]) is a contiguous 8-byte load:
//   Xp[(k>>1)*2N + 2n + (k&1)] = X[k*N + n]
// ---------------------------------------------------------------------------
__global__ void __launch_bounds__(256)
pack_pairs_kernel(const float* __restrict__ X, float* __restrict__ Xp,
                  int Ktot, int N) {
    const long idx = (long)blockIdx.x * blockDim.x + threadIdx.x;
    if (idx >= (long)Ktot * N) return;
    const int k = (int)(idx / N);
    const int n = (int)(idx % N);
    Xp[(size_t)(k >> 1) * (2 * N) + 2 * n + (k & 1)] = X[idx];
}

// ---------------------------------------------------------------------------
// C[M x N] = A[M x K] @ W[K x N] + bias[N]
// A row-major; W pre-packed in pair layout; C row-major, or pair-packed
// output when pack_out != 0 (used for the V projection).
// One wave -> 16(M) x 64(N) slab: 4 accumulators, shared A fragment,
// one-deep software pipeline. B fragments: one base pointer + immediate
// offsets (j*32 floats = 128 B) so loads stay on the GLOBAL path.
// ---------------------------------------------------------------------------
__global__ void __launch_bounds__(256)
gemm_bias_kernel(const float* __restrict__ A, const float* __restrict__ Wp,
                 const float* __restrict__ bias, float* __restrict__ C,
                 int M, int N, int K, int pack_out) {
    const int lane = threadIdx.x & 31;
    const int wave = (int)((blockIdx.x * blockDim.x + threadIdx.x) >> 5);
    const int tiles_n = N >> 6;
    const int n0 = (wave % tiles_n) << 6;
    const int m0 = (wave / tiles_n) << 4;
    if (m0 >= M) return;
    const int half = lane >> 4;   // 0: K=k,k+1   1: K=k+2,k+3
    const int lr   = lane & 15;   // M (A) / N (B,C,D) index within tile

    v8f acc[4];
    #pragma unroll
    for (int j = 0; j < 4; ++j) {
        const float bv = bias[n0 + 16 * j + lr];
        #pragma unroll
        for (int r = 0; r < 8; ++r) acc[j][r] = bv;
    }

    const float* ap = A + (size_t)(m0 + lr) * K + 2 * half;
    const float* bb = Wp + (size_t)half * (2 * N) + 2 * (n0 + lr);

    v2f a_cur = *(const v2f*)ap;
    v2f b_cur[4];
    #pragma unroll
    for (int j = 0; j < 4; ++j) b_cur[j] = *(const v2f*)(bb + 32 * j);

    #pragma unroll 2
    for (int k = 0; k < K - 4; k += 4) {
        ap += 4;
        bb += 4 * N;                             // 2 pack-rows ahead
        v2f a_nxt = *(const v2f*)ap;
        v2f b_nxt[4];
        #pragma unroll
        for (int j = 0; j < 4; ++j) b_nxt[j] = *(const v2f*)(bb + 32 * j);
        #pragma unroll
        for (int j = 0; j < 4; ++j) acc[j] = wmma_f32(a_cur, b_cur[j], acc[j]);
        a_cur = a_nxt;
        #pragma unroll
        for (int j = 0; j < 4; ++j) b_cur[j] = b_nxt[j];
    }
    #pragma unroll
    for (int j = 0; j < 4; ++j) acc[j] = wmma_f32(a_cur, b_cur[j], acc[j]);

    if (pack_out) {
        #pragma unroll
        for (int j = 0; j < 4; ++j)
            #pragma unroll
            for (int r = 0; r < 8; ++r) {
                const int row = m0 + r + 8 * half;
                const int col = n0 + 16 * j + lr;
                C[(size_t)(row >> 1) * (2 * N) + 2 * col + (row & 1)] = acc[j][r];
            }
    } else {
        #pragma unroll
        for (int j = 0; j < 4; ++j)
            #pragma unroll
            for (int r = 0; r < 8; ++r)
                C[(size_t)(m0 + r + 8 * half) * N + n0 + 16 * j + lr] = acc[j][r];
    }
}

// ---------------------------------------------------------------------------
// scores[bh, m, n] = scale * sum_d q[b, m, h*64+d] * k[b, n, h*64+d]
// B operand is K^T -> both fragment loads are naturally contiguous b64.
// Fragment j sits 16 rows (16*D_MODEL floats = 64 KB, fits imm offset)
// from the base. One wave -> 16 x 64 tile, pipelined over 16 k-steps.
// ---------------------------------------------------------------------------
__global__ void __launch_bounds__(256)
scores_kernel(const float* __restrict__ q, const float* __restrict__ kmat,
              float* __restrict__ attn) {
    const int lane = threadIdx.x & 31;
    int wave = (int)((blockIdx.x * blockDim.x + threadIdx.x) >> 5);
    const int tn = wave & 31;  wave >>= 5;    // SEQ/64 = 32 column slabs
    const int tm = wave & 127; wave >>= 7;    // SEQ/16 = 128 row tiles
    const int bh = wave;                      // 0..31
    const int b = bh >> 4, h = bh & 15;
    const int m0 = tm << 4, n0 = tn << 6;
    const int half = lane >> 4, lr = lane & 15;
    const float scale = 0.125f;               // 1/sqrt(64)

    v8f acc[4];
    #pragma unroll
    for (int j = 0; j < 4; ++j)
        #pragma unroll
        for (int r = 0; r < 8; ++r) acc[j][r] = 0.0f;

    const float* ap = q + (size_t)(b * SEQ + m0 + lr) * D_MODEL + h * D_K + 2 * half;
    const float* bb = kmat + (size_t)(b * SEQ + n0 + lr) * D_MODEL + h * D_K + 2 * half;

    v2f a_cur = *(const v2f*)ap;
    v2f b_cur[4];
    #pragma unroll
    for (int j = 0; j < 4; ++j) b_cur[j] = *(const v2f*)(bb + j * 16 * D_MODEL);

    #pragma unroll 2
    for (int k = 0; k < D_K - 4; k += 4) {
        ap += 4;
        bb += 4;
        v2f a_nxt = *(const v2f*)ap;
        v2f b_nxt[4];
        #pragma unroll
        for (int j = 0; j < 4; ++j) b_nxt[j] = *(const v2f*)(bb + j * 16 * D_MODEL);
        #pragma unroll
        for (int j = 0; j < 4; ++j) acc[j] = wmma_f32(a_cur, b_cur[j], acc[j]);
        a_cur = a_nxt;
        #pragma unroll
        for (int j = 0; j < 4; ++j) b_cur[j] = b_nxt[j];
    }
    #pragma unroll
    for (int j = 0; j < 4; ++j) acc[j] = wmma_f32(a_cur, b_cur[j], acc[j]);

    float* out = attn + ((size_t)bh * SEQ + m0) * SEQ + n0;
    #pragma unroll
    for (int j = 0; j < 4; ++j)
        #pragma unroll
        for (int r = 0; r < 8; ++r)
            out[(size_t)(r + 8 * half) * SEQ + 16 * j + lr] = acc[j][r] * scale;
}

// ---------------------------------------------------------------------------
// In-place row softmax over 2048 columns. One 256-thread block (8 waves) per
// row: wave32 shuffle reduction + LDS cross-wave combine.
// ---------------------------------------------------------------------------
__global__ void __launch_bounds__(256)
softmax_kernel(float* __restrict__ attn) {
    float* p = attn + (size_t)blockIdx.x * SEQ;
    const int t = threadIdx.x;

    float vals[8];
    float m = -3.402823466e+38f;
    #pragma unroll
    for (int i = 0; i < 8; ++i) {
        vals[i] = p[t + 256 * i];
        m = fmaxf(m, vals[i]);
    }
    #pragma unroll
    for (int off = 16; off > 0; off >>= 1) m = fmaxf(m, __shfl_xor(m, off));

    __shared__ float smax[8];
    __shared__ float ssum[8];
    const int wid = t >> 5;
    if ((t & 31) == 0) smax[wid] = m;
    __syncthreads();
    float bm = smax[0];
    #pragma unroll
    for (int w = 1; w < 8; ++w) bm = fmaxf(bm, smax[w]);

    float s = 0.0f;
    #pragma unroll
    for (int i = 0; i < 8; ++i) {
        vals[i] = __expf(vals[i] - bm);
        s += vals[i];
    }
    #pragma unroll
    for (int off = 16; off > 0; off >>= 1) s += __shfl_xor(s, off);
    if ((t & 31) == 0) ssum[wid] = s;
    __syncthreads();
    float bs = 0.0f;
    #pragma unroll
    for (int w = 0; w < 8; ++w) bs += ssum[w];

    const float inv = 1.0f / bs;
    #pragma unroll
    for (int i = 0; i < 8; ++i) p[t + 256 * i] = vals[i] * inv;
}

// ---------------------------------------------------------------------------
// ctx[b, m, h*64+n] = sum_k attn[bh, m, k] * v[b, k, h*64+n]
// A (attn) fragment loads contiguous; V is pair-packed by the projection
// GEMM so B fragments are single b64 loads at base + j*128 B.
// One wave -> 16 rows x 64 cols; K = 2048, pipelined.
// ---------------------------------------------------------------------------
__global__ void __launch_bounds__(256)
context_kernel(const float* __restrict__ attn, const float* __restrict__ vp,
               float* __restrict__ ctx) {
    const int lane = threadIdx.x & 31;
    int wave = (int)((blockIdx.x * blockDim.x + threadIdx.x) >> 5);
    const int tm = wave & 127; wave >>= 7;    // SEQ/16 = 128
    const int bh = wave;                      // 0..31
    const int b = bh >> 4, h = bh & 15;
    const int m0 = tm << 4;
    const int half = lane >> 4, lr = lane & 15;

    v8f acc[4];
    #pragma unroll
    for (int j = 0; j < 4; ++j)
        #pragma unroll
        for (int r = 0; r < 8; ++r) acc[j][r] = 0.0f;

    const float* ap = attn + ((size_t)bh * SEQ + m0 + lr) * SEQ + 2 * half;
    // packed V: pack-row p covers sequence rows 2p,2p+1; 2*D_MODEL floats/row
    const float* bb = vp + (size_t)((b * SEQ) >> 1) * (2 * D_MODEL)
                         + (size_t)half * (2 * D_MODEL)
                         + 2 * (h * D_K + lr);

    v2f a_cur = *(const v2f*)ap;
    v2f b_cur[4];
    #pragma unroll
    for (int j = 0; j < 4; ++j) b_cur[j] = *(const v2f*)(bb + 32 * j);

    #pragma unroll 2
    for (int k = 0; k < SEQ - 4; k += 4) {
        ap += 4;
        bb += 4 * D_MODEL;                       // 2 pack-rows ahead
        v2f a_nxt = *(const v2f*)ap;
        v2f b_nxt[4];
        #pragma unroll
        for (int j = 0; j < 4; ++j) b_nxt[j] = *(const v2f*)(bb + 32 * j);
        #pragma unroll
        for (int j = 0; j < 4; ++j) acc[j] = wmma_f32(a_cur, b_cur[j], acc[j]);
        a_cur = a_nxt;
        #pragma unroll
        for (int j = 0; j < 4; ++j) b_cur[j] = b_nxt[j];
    }
    #pragma unroll
    for (int j = 0; j < 4; ++j) acc[j] = wmma_f32(a_cur, b_cur[j], acc[j]);

    #pragma unroll
    for (int j = 0; j < 4; ++j)
        #pragma unroll
        for (int r = 0; r < 8; ++r)
            ctx[(size_t)(b * SEQ + m0 + r + 8 * half) * D_MODEL
                + h * D_K + 16 * j + lr] = acc[j][r];
}

// ---------------------------------------------------------------------------
extern "C" void kernel_launch(void* const* d_in, const int* in_sizes, int n_in,
                              void* d_out, int out_size, void* d_ws, size_t ws_size,
                              hipStream_t stream) {
    const float* Q   = (const float*)d_in[0];
    const float* K   = (const float*)d_in[1];
    const float* V   = (const float*)d_in[2];
    const float* Wq  = (const float*)d_in[3];
    const float* bq  = (const float*)d_in[4];
    const float* Wk  = (const float*)d_in[5];
    const float* bk  = (const float*)d_in[6];
    const float* Wv  = (const float*)d_in[7];
    const float* bv  = (const float*)d_in[8];
    const float* Wfc = (const float*)d_in[9];
    const float* bfc = (const float*)d_in[10];

    const size_t M = (size_t)BATCH * SEQ;           // 4096 rows
    const size_t proj_elems = M * D_MODEL;          // 4.19M floats = 16 MB
    const size_t w_elems    = (size_t)D_MODEL * D_MODEL; // 1M floats = 4 MB

    float* q    = (float*)d_ws;                     // [4096,1024]
    float* kk   = q    + proj_elems;                // [4096,1024]
    float* vpk  = kk   + proj_elems;                // pair-packed V, 16 MB
    float* ctx  = vpk  + proj_elems;                // [4096,1024]
    float* Wqp  = ctx  + proj_elems;                // packed weights, 4 MB each
    float* Wkp  = Wqp  + w_elems;
    float* Wvp  = Wkp  + w_elems;
    float* Wfcp = Wvp  + w_elems;                   // total ws: 80 MB

    float* out  = (float*)d_out;                    // [B,S,D]  = 4.19M floats
    float* attn = out + proj_elems;                 // [B,H,S,S] = 134.2M floats

    dim3 blk(256);
    const int pack_blocks  = (int)(w_elems / 256);                           // 4096
    const int gemm_blocks  = (int)((M / 16) * (D_MODEL / 64) / 8);           // 512
    const int score_blocks = BATCH * NUM_HEADS * (SEQ / 16) * (SEQ / 64) / 8; // 16384
    const int ctx_blocks   = BATCH * NUM_HEADS * (SEQ / 16) / 8;             // 512
    const int soft_blocks  = BATCH * NUM_HEADS * SEQ;                        // 65536

    pack_pairs_kernel<<<pack_blocks, blk, 0, stream>>>(Wq,  Wqp,  D_MODEL, D_MODEL);
    pack_pairs_kernel<<<pack_blocks, blk, 0, stream>>>(Wk,  Wkp,  D_MODEL, D_MODEL);
    pack_pairs_kernel<<<pack_blocks, blk, 0, stream>>>(Wv,  Wvp,  D_MODEL, D_MODEL);
    pack_pairs_kernel<<<pack_blocks, blk, 0, stream>>>(Wfc, Wfcp, D_MODEL, D_MODEL);

    gemm_bias_kernel<<<gemm_blocks, blk, 0, stream>>>(Q, Wqp, bq, q,   (int)M, D_MODEL, D_MODEL, 0);
    gemm_bias_kernel<<<gemm_blocks, blk, 0, stream>>>(K, Wkp, bk, kk,  (int)M, D_MODEL, D_MODEL, 0);
    gemm_bias_kernel<<<gemm_blocks, blk, 0, stream>>>(V, Wvp, bv, vpk, (int)M, D_MODEL, D_MODEL, 1);

    scores_kernel<<<score_blocks, blk, 0, stream>>>(q, kk, attn);
    softmax_kernel<<<soft_blocks, blk, 0, stream>>>(attn);
    context_kernel<<<ctx_blocks, blk, 0, stream>>>(attn, vpk, ctx);

    gemm_bias_kernel<<<gemm_blocks, blk, 0, stream>>>(ctx, Wfcp, bfc, out, (int)M, D_MODEL, D_MODEL, 0);
}